// HybridLoss_23476291240292
// MI455X (gfx1250) — compile-verified
//
#include <hip/hip_runtime.h>
#include <hip/hip_bf16.h>

#define NPTS   8192
#define EPS    0.05f
#define ALPHA  0.2f
#define NITERS 50
#define LN2    0.6931471805599453f
#define LOG2E  1.4426950408889634f

typedef float v2f __attribute__((ext_vector_type(2)));
typedef float v8f __attribute__((ext_vector_type(8)));

// ---------------------------------------------------------------------------
// Block-wide sum reduction (blockDim.x == 256 -> 8 waves of 32)
// Result valid in thread 0.
// ---------------------------------------------------------------------------
__device__ inline float block_sum(float v) {
    __shared__ float red[8];
    const int lane = threadIdx.x & 31;
    const int w    = threadIdx.x >> 5;
    #pragma unroll
    for (int off = 16; off > 0; off >>= 1) v += __shfl_xor(v, off, 32);
    if (lane == 0) red[w] = v;
    __syncthreads();
    float out = 0.0f;
    if (w == 0) {
        out = (lane < 8) ? red[lane] : 0.0f;
        #pragma unroll
        for (int off = 4; off > 0; off >>= 1) out += __shfl_xor(out, off, 8);
    }
    __syncthreads();
    return out;
}

// ---------------------------------------------------------------------------
// Prep: strip w, compute squared norms, zero-init the 6 dual arrays.
// ---------------------------------------------------------------------------
__global__ void prep_kernel(const float* __restrict__ pred, const float* __restrict__ gt,
                            float* __restrict__ xw, float* __restrict__ yw,
                            float* __restrict__ nx, float* __restrict__ ny,
                            float* __restrict__ FG) {
    const int i = blockIdx.x * blockDim.x + threadIdx.x;   // 0 .. 2*NPTS-1
    if (i < NPTS) {
        const float a = pred[i * 4 + 0], b = pred[i * 4 + 1], c = pred[i * 4 + 2];
        xw[i * 4 + 0] = a; xw[i * 4 + 1] = b; xw[i * 4 + 2] = c; xw[i * 4 + 3] = 0.0f;
        nx[i] = a * a + b * b + c * c;
    } else {
        const int j = i - NPTS;
        const float a = gt[j * 4 + 0], b = gt[j * 4 + 1], c = gt[j * 4 + 2];
        yw[j * 4 + 0] = a; yw[j * 4 + 1] = b; yw[j * 4 + 2] = c; yw[j * 4 + 3] = 0.0f;
        ny[j] = a * a + b * b + c * c;
    }
    #pragma unroll
    for (int k = 0; k < 3; ++k) FG[k * 2 * NPTS + i] = 0.0f;
}

// ---------------------------------------------------------------------------
// One Sinkhorn half-step. Each BLOCK owns one 16-row group; its 8 waves split
// the 8192 columns (1024 each), each wave producing per-row (max, scaled-sum)
// logsumexp state in base-2 domain; states merge across lanes via shuffles
// and across waves via LDS.
//
// dual_out[row] = -eps*ln2*( m2 + log2(s2) - 13 )      (log(1/8192) folded in)
//
// C recomputed on the fly: V_WMMA_F32_16X16X4_F32 gives a 16x16 tile of dot
// products per instruction (K=4, w component zeroed), then norms + raw sqrt.
// ---------------------------------------------------------------------------
__global__ void sinkhorn_pass_kernel(const float* __restrict__ X, const float* __restrict__ Y,
                                     const float* __restrict__ NX, const float* __restrict__ NY,
                                     float* __restrict__ FG, int phase) {
    const int prob = blockIdx.y;
    const float *apts, *anorm, *bpts, *bnorm;
    float* f = FG + prob * 2 * NPTS;
    float* g = f + NPTS;

    if (prob == 0) {
        if (phase == 0) { apts = X; anorm = NX; bpts = Y; bnorm = NY; }  // f over pred rows
        else            { apts = Y; anorm = NY; bpts = X; bnorm = NX; }  // g over gt rows (C^T)
    } else if (prob == 1) {
        apts = X; anorm = NX; bpts = X; bnorm = NX;
    } else {
        apts = Y; anorm = NY; bpts = Y; bnorm = NY;
    }
    const float* din  = (phase == 0) ? g : f;
    float*       dout = (phase == 0) ? f : g;

    const int row0 = blockIdx.x * 16;
    const int wv   = threadIdx.x >> 5;       // wave id 0..7 -> column chunk
    const int lane = threadIdx.x & 31;
    const int half = lane >> 4;              // 0: K=0..1 / rows M+0..7, 1: K=2..3 / rows M+8..15
    const int idx  = lane & 15;              // column within tile / row within A tile

    __shared__ float lm[8][16];
    __shared__ float ls[8][16];

    // A operand: 16x4 f32, lane idx holds row (row0+idx), K = 2*half .. 2*half+1
    v2f Av;
    Av.x = apts[(row0 + idx) * 4 + 2 * half + 0];
    Av.y = apts[(row0 + idx) * 4 + 2 * half + 1];

    float na[8];
    #pragma unroll
    for (int r = 0; r < 8; ++r) na[r] = anorm[row0 + r + 8 * half];

    float m[8], s[8];
    #pragma unroll
    for (int r = 0; r < 8; ++r) { m[r] = -3.0e38f; s[r] = 0.0f; }

    const float K2 = LOG2E / EPS;            // base-2 scaling of (g - C)/eps

    const int cbase = wv * (NPTS / 8);       // 1024 columns per wave
    for (int t = 0; t < (NPTS / 8); t += 16) {
        const int c0 = cbase + t;
        // B operand: 4x16 f32 = (b tile)^T, lane idx holds column (c0+idx)
        v2f Bv;
        Bv.x = bpts[(c0 + idx) * 4 + 2 * half + 0];
        Bv.y = bpts[(c0 + idx) * 4 + 2 * half + 1];

        v8f acc = {};
        acc = __builtin_amdgcn_wmma_f32_16x16x4_f32(
            false, Av, false, Bv, (short)0, acc, false, false);

        const float nb  = bnorm[c0 + idx];
        const float gjK = din[c0 + idx] * K2;

        #pragma unroll
        for (int r = 0; r < 8; ++r) {
            float d2 = na[r] + nb - 2.0f * acc[r];
            d2 = fmaxf(d2, 0.0f) + 1e-12f;
            const float cost = __builtin_amdgcn_sqrtf(d2);   // raw v_sqrt_f32
            const float w = fmaf(cost, -K2, gjK);            // (g - C) * log2e/eps
            // online logsumexp (base 2)
            const float mo = m[r];
            const float mn = fmaxf(mo, w);
            s[r] = fmaf(s[r], __builtin_amdgcn_exp2f(mo - mn),
                        __builtin_amdgcn_exp2f(w - mn));
            m[r] = mn;
        }
    }

    // Merge (m,s) across the 16 lanes of each half-wave.
    // Slot r: lanes 0..15 hold row (row0+r), lanes 16..31 hold row (row0+r+8).
    #pragma unroll
    for (int r = 0; r < 8; ++r) {
        float mm = m[r], ss = s[r];
        #pragma unroll
        for (int mask = 1; mask < 16; mask <<= 1) {
            const float mo = __shfl_xor(mm, mask, 16);
            const float so = __shfl_xor(ss, mask, 16);
            const float mn = fmaxf(mm, mo);
            ss = fmaf(ss, __builtin_amdgcn_exp2f(mm - mn),
                      so * __builtin_amdgcn_exp2f(mo - mn));
            mm = mn;
        }
        m[r] = mm; s[r] = ss;
    }

    if (idx == 0) {
        #pragma unroll
        for (int r = 0; r < 8; ++r) {
            lm[wv][r + 8 * half] = m[r];
            ls[wv][r + 8 * half] = s[r];
        }
    }
    __syncthreads();

    // Merge the 8 wave-partials per row and emit.
    if (threadIdx.x < 16) {
        const int rl = threadIdx.x;
        float mm = lm[0][rl], ss = ls[0][rl];
        #pragma unroll
        for (int w2 = 1; w2 < 8; ++w2) {
            const float mo = lm[w2][rl], so = ls[w2][rl];
            const float mn = fmaxf(mm, mo);
            ss = fmaf(ss, __builtin_amdgcn_exp2f(mm - mn),
                      so * __builtin_amdgcn_exp2f(mo - mn));
            mm = mn;
        }
        // log(1/NPTS) = -13*ln2 folds into the base-2 result
        dout[row0 + rl] = -EPS * LN2 * (mm + __builtin_amdgcn_logf(ss) - 13.0f);
    }
}

// ---------------------------------------------------------------------------
// Chamfer: for each gt row i, min_j d2(gt_i, pred_j); block-partial sums.
// ---------------------------------------------------------------------------
__global__ void chamfer_kernel(const float* __restrict__ xw, const float* __restrict__ yw,
                               const float* __restrict__ nx, const float* __restrict__ ny,
                               float* __restrict__ partial) {
    const int i = blockIdx.x * blockDim.x + threadIdx.x;
    const float gx = yw[i * 4 + 0], gy = yw[i * 4 + 1], gz = yw[i * 4 + 2];
    const float nyi = ny[i];
    float best = 3.0e38f;
    for (int j = 0; j < NPTS; ++j) {
        const float dot = gx * xw[j * 4 + 0] + gy * xw[j * 4 + 1] + gz * xw[j * 4 + 2];
        const float d2 = nyi + nx[j] - 2.0f * dot;
        best = fminf(best, fmaxf(d2, 0.0f));
    }
    const float bs = block_sum(best);
    if (threadIdx.x == 0) partial[blockIdx.x] = bs;
}

// ---------------------------------------------------------------------------
// Finalize: sum duals -> OT values, sum chamfer partials, combine.
// ---------------------------------------------------------------------------
__global__ void finalize_kernel(const float* __restrict__ FG, const float* __restrict__ partial,
                                int n_partial, float* __restrict__ out) {
    float sums[6];
    #pragma unroll
    for (int k = 0; k < 6; ++k) {
        float v = 0.0f;
        for (int i = threadIdx.x; i < NPTS; i += blockDim.x) v += FG[k * NPTS + i];
        sums[k] = block_sum(v);
    }
    float v = 0.0f;
    for (int i = threadIdx.x; i < n_partial; i += blockDim.x) v += partial[i];
    const float chamfer_sum = block_sum(v);

    if (threadIdx.x == 0) {
        const float inv_n = 1.0f / (float)NPTS;
        const float ot_xy = (sums[0] + sums[1]) * inv_n;
        const float ot_xx = (sums[2] + sums[3]) * inv_n;
        const float ot_yy = (sums[4] + sums[5]) * inv_n;
        const float emd = ot_xy - 0.5f * ot_xx - 0.5f * ot_yy;
        const float chamfer = chamfer_sum * inv_n;
        out[0] = ALPHA * chamfer + (1.0f - ALPHA) * emd;
    }
}

// ---------------------------------------------------------------------------
// Launch
// ---------------------------------------------------------------------------
extern "C" void kernel_launch(void* const* d_in, const int* in_sizes, int n_in,
                              void* d_out, int out_size, void* d_ws, size_t ws_size,
                              hipStream_t stream) {
    (void)in_sizes; (void)n_in; (void)out_size; (void)ws_size;

    const float* pred = (const float*)d_in[0];   // 8192 x 4
    const float* gt   = (const float*)d_in[1];   // 8192 x 4

    float* ws = (float*)d_ws;
    float* xw      = ws;                 // 4*NPTS : pred xyz, w=0
    float* yw      = xw + 4 * NPTS;      // 4*NPTS : gt xyz, w=0
    float* nx      = yw + 4 * NPTS;      // NPTS
    float* ny      = nx + NPTS;          // NPTS
    float* FG      = ny + NPTS;          // 6*NPTS : f_xy,g_xy,f_xx,g_xx,f_yy,g_yy
    float* partial = FG + 6 * NPTS;      // 32

    prep_kernel<<<(2 * NPTS) / 256, 256, 0, stream>>>(pred, gt, xw, yw, nx, ny, FG);

    chamfer_kernel<<<NPTS / 256, 256, 0, stream>>>(xw, yw, nx, ny, partial);

    const dim3 sgrid(NPTS / 16, 3);      // one block per 16-row group, 3 OT problems
    for (int it = 0; it < NITERS; ++it) {
        sinkhorn_pass_kernel<<<sgrid, 256, 0, stream>>>(xw, yw, nx, ny, FG, 0);
        sinkhorn_pass_kernel<<<sgrid, 256, 0, stream>>>(xw, yw, nx, ny, FG, 1);
    }

    finalize_kernel<<<1, 256, 0, stream>>>(FG, partial, NPTS / 256, (float*)d_out);
}